// SNN_44959717654654
// MI455X (gfx1250) — compile-verified
//
#include <hip/hip_runtime.h>
#include <hip/hip_bf16.h>

#define BATCH 8192
#define C1 16
#define C2 32
#define POOL1 12
#define FEAT 512
#define NOUT 950
#define NSTEP 10
#define NPER 95
#define BETA 0.95f
#define THRESH 1.0f

typedef __attribute__((ext_vector_type(2))) float v2f;
typedef __attribute__((ext_vector_type(8))) float v8f;
typedef __attribute__((ext_vector_type(2))) int   v2i;

__device__ __forceinline__ v8f wmma4(v2f a, v2f b, v8f c) {
  return __builtin_amdgcn_wmma_f32_16x16x4_f32(
      /*neg_a=*/false, a, /*neg_b=*/false, b,
      /*c_mod=*/(short)0, c, /*reuse_a=*/false, /*reuse_b=*/false);
}

// ---------------- conv1 (5x5, 1->16) + ReLU + 2x2 maxpool ----------------
// one thread per pooled output element of [B,16,12,12]
__global__ __launch_bounds__(256) void conv1_pool_kernel(
    const float* __restrict__ x, const float* __restrict__ w1,
    const float* __restrict__ b1, float* __restrict__ h1) {
  int idx = blockIdx.x * blockDim.x + threadIdx.x;
  const int N = BATCH * C1 * POOL1 * POOL1;
  if (idx >= N) return;
  int px = idx % POOL1;
  int py = (idx / POOL1) % POOL1;
  int oc = (idx / (POOL1 * POOL1)) % C1;
  int b  = idx / (POOL1 * POOL1 * C1);

  const float* xb = x + (size_t)b * 28 * 28;
  const float* w  = w1 + oc * 25;

  float p[6][6];
#pragma unroll
  for (int i = 0; i < 6; ++i)
#pragma unroll
    for (int j = 0; j < 6; ++j)
      p[i][j] = xb[(2 * py + i) * 28 + (2 * px + j)];

  float bias = b1[oc];
  float m = 0.0f;  // all candidates are post-ReLU (>=0)
#pragma unroll
  for (int dy = 0; dy < 2; ++dy)
#pragma unroll
    for (int dx = 0; dx < 2; ++dx) {
      float acc = bias;
#pragma unroll
      for (int ky = 0; ky < 5; ++ky)
#pragma unroll
        for (int kx = 0; kx < 5; ++kx)
          acc = fmaf(p[dy + ky][dx + kx], w[ky * 5 + kx], acc);
      acc = fmaxf(acc, 0.0f);
      m = fmaxf(m, acc);
    }
  h1[idx] = m;
}

// ------- conv2 as implicit GEMM on WMMA + fused ReLU + 2x2 maxpool --------
// One wave = one (batch b, pooled row py) pair, all 32 output channels.
// GEMM: M = 16 conv positions {oy in {2py,2py+1}} x {ox in 0..7}, N = 32 oc
// (2 N-tiles), K = 400 = ic*25 + ky*5 + kx.  w2 [32,16,5,5] == row-major
// [32,400] so B-frags are contiguous float2 loads.  A-frag gathers use an
// LDS table of (off(kk), off(kk+1)) pairs, off(kk) = ic*144 + ky*12 + kx.
// Pool: partners are acc[2px]/acc[2px+1] in-lane and the hi-half via
// shfl_xor(16) (D: lane l, vgpr r -> m = r + 8*(l>=16), m = (oy-2py)*8+ox).
__global__ __launch_bounds__(128) void conv2_wmma_pool_kernel(
    const float* __restrict__ h1, const float* __restrict__ w2,
    const float* __restrict__ b2, float* __restrict__ h2) {
  __shared__ v2i offs2[200];
  for (int j = threadIdx.x; j < 200; j += blockDim.x) {
    int kk = 2 * j;
    int ic0 = kk / 25,      r0 = kk % 25;
    int ic1 = (kk + 1) / 25, r1 = (kk + 1) % 25;
    v2i o;
    o.x = ic0 * 144 + (r0 / 5) * 12 + (r0 % 5);
    o.y = ic1 * 144 + (r1 / 5) * 12 + (r1 % 5);
    offs2[j] = o;
  }
  __syncthreads();

  int wid  = blockIdx.x * (blockDim.x >> 5) + (threadIdx.x >> 5);
  int lane = threadIdx.x & 31;
  int hi = lane >> 4;
  int lid = lane & 15;
  int b  = wid >> 2;
  int py = wid & 3;

  // A row m = lid -> conv position oy = 2*py + (lid>>3), ox = lid&7
  int posoff = (2 * py + (lid >> 3)) * POOL1 + (lid & 7);
  const float* abase = h1 + (size_t)b * (C1 * POOL1 * POOL1) + posoff;
  const float* brow  = w2 + (size_t)lid * 400 + 2 * hi;  // oc=lid; +6400 -> oc=16+lid

  v8f acc0 = {0.f, 0.f, 0.f, 0.f, 0.f, 0.f, 0.f, 0.f};
  v8f acc1 = {0.f, 0.f, 0.f, 0.f, 0.f, 0.f, 0.f, 0.f};

#pragma unroll 2
  for (int k = 0; k < 400; k += 4) {
    v2i o = offs2[(k >> 1) + hi];  // offsets for kk = k+2*hi, kk+1
    v2f a;
    a.x = abase[o.x];
    a.y = abase[o.y];
    v2f bv0 = *(const v2f*)(brow + k);
    v2f bv1 = *(const v2f*)(brow + 6400 + k);
    acc0 = wmma4(a, bv0, acc0);
    acc1 = wmma4(a, bv1, acc1);
  }

  // epilogue: +bias, ReLU, 2x2 maxpool, store h2[b][oc][py][px] (flat b*512+oc*16+py*4+px)
  float bias0 = b2[lid];
  float bias1 = b2[16 + lid];
  float* dst = h2 + (size_t)b * FEAT + py * 4;
#pragma unroll
  for (int px = 0; px < 4; ++px) {
    float v0 = fmaxf(fmaxf(acc0[2 * px] + bias0, 0.0f),
                     fmaxf(acc0[2 * px + 1] + bias0, 0.0f));
    float v1 = fmaxf(fmaxf(acc1[2 * px] + bias1, 0.0f),
                     fmaxf(acc1[2 * px + 1] + bias1, 0.0f));
    v0 = fmaxf(v0, __shfl_xor(v0, 16, 32));  // combine oy halves
    v1 = fmaxf(v1, __shfl_xor(v1, 16, 32));
    if (lane < 16) {
      dst[lid * 16 + px]        = v0;
      dst[(16 + lid) * 16 + px] = v1;
    }
  }
}

// -------- FC: out = h2 @ wf^T + bf, 64x32 tile per wave on WMMA f32 --------
// 4 M-tiles x 2 N-tiles = 8 accumulators; per k-step 4 A + 2 B float2 loads
// feed 8 WMMAs (load:wmma = 0.75).  3840 waves total.
__global__ __launch_bounds__(128) void fc_wmma_kernel(
    const float* __restrict__ h2, const float* __restrict__ wf,
    const float* __restrict__ bf, float* __restrict__ fcout) {
  const int NB = 30;  // N-blocks of 32 (covers 960; cols >= 950 masked at store)
  int wave = blockIdx.x * (blockDim.x >> 5) + (threadIdx.x >> 5);
  int lane = threadIdx.x & 31;
  int hi = lane >> 4;
  int lid = lane & 15;
  int mb = wave / NB;  // 0..127  (64 rows each)
  int nb = wave % NB;  // 0..29   (32 cols each)

  int col0 = nb * 32 + lid;
  int col1 = col0 + 16;
  int c0 = col0 < NOUT ? col0 : NOUT - 1;  // clamp: junk cols never stored
  int c1 = col1 < NOUT ? col1 : NOUT - 1;

  const float* arow = h2 + ((size_t)mb * 64 + lid) * FEAT + 2 * hi;
  const float* br0  = wf + (size_t)c0 * FEAT + 2 * hi;
  const float* br1  = wf + (size_t)c1 * FEAT + 2 * hi;

  v8f acc[4][2] = {};
#pragma unroll 2
  for (int k = 0; k < FEAT; k += 4) {
    v2f bv0 = *(const v2f*)(br0 + k);
    v2f bv1 = *(const v2f*)(br1 + k);
#pragma unroll
    for (int mm = 0; mm < 4; ++mm) {
      v2f a = *(const v2f*)(arow + mm * (16 * FEAT) + k);
      acc[mm][0] = wmma4(a, bv0, acc[mm][0]);
      acc[mm][1] = wmma4(a, bv1, acc[mm][1]);
    }
  }

#pragma unroll
  for (int nt = 0; nt < 2; ++nt) {
    int col = nt ? col1 : col0;
    if (col < NOUT) {
      float bias = bf[col];
#pragma unroll
      for (int mm = 0; mm < 4; ++mm) {
        size_t m0 = (size_t)mb * 64 + mm * 16 + 8 * hi;
#pragma unroll
        for (int r = 0; r < 8; ++r)
          fcout[(m0 + r) * NOUT + col] = acc[mm][nt][r] + bias;
      }
    }
  }
}

// ---------------- LIF scan: [8192,950] -> spk/mem [10,8192,95] -------------
__global__ __launch_bounds__(256) void lif_kernel(
    const float* __restrict__ fcout, float* __restrict__ out) {
  int idx = blockIdx.x * blockDim.x + threadIdx.x;
  const int N = BATCH * NPER;
  if (idx >= N) return;
  int n = idx % NPER;
  int b = idx / NPER;
  const size_t TOT = (size_t)NSTEP * BATCH * NPER;
  const float* src = fcout + (size_t)b * NOUT + n;
  float mem = 0.0f;
#pragma unroll
  for (int t = 0; t < NSTEP; ++t) {
    float inp   = src[t * NPER];
    float reset = mem > THRESH ? 1.0f : 0.0f;
    mem = BETA * mem + inp - reset * THRESH;
    float spk = mem > THRESH ? 1.0f : 0.0f;
    size_t o = (size_t)t * (BATCH * NPER) + (size_t)b * NPER + n;
    out[o]       = spk;
    out[TOT + o] = mem;
  }
}

extern "C" void kernel_launch(void* const* d_in, const int* in_sizes, int n_in,
                              void* d_out, int out_size, void* d_ws, size_t ws_size,
                              hipStream_t stream) {
  const float* x  = (const float*)d_in[0];
  const float* w1 = (const float*)d_in[1];
  const float* b1 = (const float*)d_in[2];
  const float* w2 = (const float*)d_in[3];
  const float* b2 = (const float*)d_in[4];
  const float* wf = (const float*)d_in[5];
  const float* bf = (const float*)d_in[6];
  float* out = (float*)d_out;

  // workspace layout
  const size_t N_H1 = (size_t)BATCH * C1 * POOL1 * POOL1;  // 18,874,368 floats
  const size_t N_H2 = (size_t)BATCH * FEAT;                //  4,194,304 floats
  float* h1    = (float*)d_ws;
  float* h2    = h1 + N_H1;
  float* fcout = h2 + N_H2;

  {
    int n = BATCH * C1 * POOL1 * POOL1;
    conv1_pool_kernel<<<(n + 255) / 256, 256, 0, stream>>>(x, w1, b1, h1);
  }
  {
    int waves = BATCH * 4;  // (b, py) pairs
    conv2_wmma_pool_kernel<<<waves / 4, 128, 0, stream>>>(h1, w2, b2, h2);
  }
  {
    int waves = (BATCH / 64) * 30;  // 3840
    fc_wmma_kernel<<<waves / 4, 128, 0, stream>>>(h2, wf, bf, fcout);
  }
  {
    int n = BATCH * NPER;
    lif_kernel<<<(n + 255) / 256, 256, 0, stream>>>(fcout, out);
  }
}